// encembed_scamp_15994458211145
// MI455X (gfx1250) — compile-verified
//
#include <hip/hip_runtime.h>

typedef __attribute__((ext_vector_type(2))) float v2f;
typedef __attribute__((ext_vector_type(8))) float v8f;

namespace {
constexpr int B_ = 8, S_ = 2048, C_ = 8, M_ = 64, K_ = 8, D_ = 512;
constexpr int EXCL_ = M_ / 4;       // 16
constexpr int N_ = S_ - M_ + 1;     // 1985
constexpr int TILE_ = 128;
constexpr int NT_ = 16;             // tiles per dim (16*128 = 2048 >= 1985)
}

// ---------- helpers ----------
__device__ __forceinline__ void topk_insert8(unsigned long long* top,
                                             unsigned long long key) {
  if (key < top[7]) {
    top[7] = key;
#pragma unroll
    for (int q = 7; q > 0; --q) {
      if (top[q] < top[q - 1]) {
        unsigned long long t = top[q - 1];
        top[q - 1] = top[q];
        top[q] = t;
      }
    }
  }
}

// merge two ascending 8-lists (LDS), keep 8 smallest into dst
__device__ __forceinline__ void merge8(unsigned long long* dst,
                                       const unsigned long long* other) {
  unsigned long long out[8];
  int ia = 0, ib = 0;
#pragma unroll
  for (int t = 0; t < 8; ++t) {
    bool takeA = (ib >= 8) || (ia < 8 && dst[ia] <= other[ib]);
    out[t] = takeA ? dst[ia++] : other[ib++];
  }
#pragma unroll
  for (int t = 0; t < 8; ++t) dst[t] = out[t];
}

// ---------- stage 1: extract series, window mean/std ----------
__global__ void __launch_bounds__(256)
k_prep(const float* __restrict__ x, float* __restrict__ ts,
       float* __restrict__ mu, float* __restrict__ sig) {
  const int b = blockIdx.x;
  const int tid = threadIdx.x;
  __shared__ float ls[S_];
  const float* xb = x + (size_t)b * S_ * C_;
  for (int t = tid; t < S_; t += 256) {
    float v = xb[t * C_];  // channel 0
    ls[t] = v;
    ts[b * S_ + t] = v;
  }
  __syncthreads();
  for (int i = tid; i < N_; i += 256) {
    float s = 0.f;
    for (int m = 0; m < M_; ++m) s += ls[i + m];
    float mean = s * (1.0f / (float)M_);
    float s2 = 0.f;
    for (int m = 0; m < M_; ++m) {
      float d = ls[i + m] - mean;
      s2 += d * d;
    }
    mu[b * S_ + i] = mean;
    sig[b * S_ + i] = sqrtf(s2 * (1.0f / (float)M_));
  }
}

// epilogue for one 16x16 correlation subtile: dist + exclusion + top-8
__device__ __forceinline__ void tile_epilogue(
    const v8f& acc, int nj, int i0, int j0, int wave, int hi, int ml,
    const float* mui_r, const float* sgi_r, const float* muj_ls,
    const float* sgj_ls, unsigned long long* top) {
  const int j = j0 + nj * 16 + ml;
  const float mj = muj_ls[nj * 16 + ml];
  const float sj = sgj_ls[nj * 16 + ml];
#pragma unroll
  for (int r = 0; r < 8; ++r) {
    const int i = i0 + wave * 16 + r + 8 * hi;  // C/D layout: M = r + 8*hi
    int di = i - j;
    if (di < 0) di = -di;
    if (i < N_ && j < N_ && di >= EXCL_) {
      float corr =
          (acc[r] - 64.0f * mui_r[r] * mj) / fmaxf(64.0f * sgi_r[r] * sj, 1e-12f);
      float dist = sqrtf(fmaxf(128.0f * (1.0f - corr), 0.0f));
      unsigned long long key =
          ((unsigned long long)__float_as_uint(dist) << 32) |
          (unsigned int)(i * N_ + j);
      topk_insert8(top, key);
    }
  }
}

// ---------- stage 2: tiled WMMA correlation + fused streaming top-8 ----------
__global__ void __launch_bounds__(256)
k_scamp(const float* __restrict__ ts, const float* __restrict__ mu,
        const float* __restrict__ sig, unsigned long long* __restrict__ cand) {
  const int b = blockIdx.y;
  const int ti = blockIdx.x >> 4;
  const int tj = blockIdx.x & 15;
  const int i0 = ti * TILE_;
  const int j0 = tj * TILE_;
  const int tid = threadIdx.x;
  const int lane = tid & 31;
  const int wave = tid >> 5;
  const int hi = lane >> 4;   // half-wave selector
  const int ml = lane & 15;

  __shared__ float a_ls[TILE_ + M_];  // ts[i0 .. i0+191]
  __shared__ float b_ls[TILE_ + M_];  // ts[j0 .. j0+191]
  __shared__ float mui[TILE_], sgi[TILE_], muj[TILE_], sgj[TILE_];
  __shared__ unsigned long long km[256 * 8];

  const float* tsb = ts + (size_t)b * S_;
  for (int t = tid; t < TILE_ + M_; t += 256) {
    int ia = i0 + t;
    a_ls[t] = (ia < S_) ? tsb[ia] : 0.f;
    int ja = j0 + t;
    b_ls[t] = (ja < S_) ? tsb[ja] : 0.f;
  }
  if (tid < TILE_) {
    int ii = i0 + tid, jj = j0 + tid;
    mui[tid] = (ii < N_) ? mu[(size_t)b * S_ + ii] : 0.f;
    sgi[tid] = (ii < N_) ? sig[(size_t)b * S_ + ii] : 1.f;
    muj[tid] = (jj < N_) ? mu[(size_t)b * S_ + jj] : 0.f;
    sgj[tid] = (jj < N_) ? sig[(size_t)b * S_ + jj] : 1.f;
  }
  __syncthreads();

  unsigned long long top[8];
#pragma unroll
  for (int t = 0; t < 8; ++t) top[t] = ~0ull;

  // hoist per-thread row statistics out of the hot loop (register cache)
  float mui_r[8], sgi_r[8];
#pragma unroll
  for (int r = 0; r < 8; ++r) {
    const int il = wave * 16 + r + 8 * hi;
    mui_r[r] = mui[il];
    sgi_r[r] = sgi[il];
  }

  const int arow = wave * 16 + ml;  // A-fragment row (windows are shifted ts)
  for (int njp = 0; njp < 8; njp += 2) {  // two subtiles => two indep. chains
    v8f acc0 = {};
    v8f acc1 = {};
    const int bcol0 = njp * 16 + ml;
    const int bcol1 = bcol0 + 16;
#pragma unroll
    for (int kk = 0; kk < M_; kk += 4) {
      v2f av, bv0, bv1;
      av.x = a_ls[arow + kk + 2 * hi];        // A[m][kk+2*hi]
      av.y = a_ls[arow + kk + 2 * hi + 1];
      bv0.x = b_ls[bcol0 + kk + 2 * hi];      // B0[kk+2*hi][n]
      bv0.y = b_ls[bcol0 + kk + 2 * hi + 1];
      bv1.x = b_ls[bcol1 + kk + 2 * hi];      // B1[kk+2*hi][n]
      bv1.y = b_ls[bcol1 + kk + 2 * hi + 1];
      acc0 = __builtin_amdgcn_wmma_f32_16x16x4_f32(
          false, av, false, bv0, (short)0, acc0, false, false);
      acc1 = __builtin_amdgcn_wmma_f32_16x16x4_f32(
          false, av, false, bv1, (short)0, acc1, false, false);
    }
    tile_epilogue(acc0, njp + 0, i0, j0, wave, hi, ml, mui_r, sgi_r, muj, sgj,
                  top);
    tile_epilogue(acc1, njp + 1, i0, j0, wave, hi, ml, mui_r, sgi_r, muj, sgj,
                  top);
  }

  // tree-merge the 256 per-thread sorted-8 lists
#pragma unroll
  for (int t = 0; t < 8; ++t) km[tid * 8 + t] = top[t];
  __syncthreads();
  for (int stride = 128; stride >= 1; stride >>= 1) {
    if (tid < stride) merge8(&km[tid * 8], &km[(tid + stride) * 8]);
    __syncthreads();
  }
  if (tid < 8) cand[((size_t)b * 256 + blockIdx.x) * 8 + tid] = km[tid];
}

// ---------- stage 3: per-batch global top-8 -> patch starts ----------
__global__ void __launch_bounds__(256)
k_select(const unsigned long long* __restrict__ cand, int* __restrict__ starts) {
  const int b = blockIdx.x;
  const int tid = threadIdx.x;
  __shared__ unsigned long long km[256 * 8];
  unsigned long long top[8];
#pragma unroll
  for (int t = 0; t < 8; ++t) top[t] = ~0ull;
  const unsigned long long* cb = cand + (size_t)b * 256 * 8;
  for (int t = tid; t < 256 * 8; t += 256) topk_insert8(top, cb[t]);
#pragma unroll
  for (int t = 0; t < 8; ++t) km[tid * 8 + t] = top[t];
  __syncthreads();
  for (int stride = 128; stride >= 1; stride >>= 1) {
    if (tid < stride) merge8(&km[tid * 8], &km[(tid + stride) * 8]);
    __syncthreads();
  }
  if (tid < 8) {
    unsigned int flat = (unsigned int)(km[tid] & 0xffffffffull);
    int j = (int)(flat % (unsigned int)N_);  // flat_idx % n
    int st = j - M_ / 2;
    st = st < 0 ? 0 : st;
    st = st > (S_ - M_) ? (S_ - M_) : st;
    starts[b * 8 + tid] = st;
  }
}

// ---------- stage 4: gather patches + WMMA projection GEMM + bias ----------
__global__ void __launch_bounds__(32)
k_outgemm(const float* __restrict__ x, const float* __restrict__ W,
          const float* __restrict__ bias, const int* __restrict__ starts,
          float* __restrict__ out) {
  const int b = blockIdx.z;
  const int rt = blockIdx.y;  // 0..3  (rows = C*K = 64)
  const int ct = blockIdx.x;  // 0..15 (each wave covers 32 of D=512 cols)
  const int lane = threadIdx.x & 31;
  const int hi = lane >> 4;
  const int ml = lane & 15;

  const int row = rt * 16 + ml;  // row = c*8 + k
  const int ch = row >> 3;
  const int kn = row & 7;
  const int st = starts[b * 8 + kn];
  const float* xb = x + (size_t)b * S_ * C_;
  const int d0 = ct * 32 + ml;
  const int d1 = d0 + 16;

  v8f acc0 = {};
  v8f acc1 = {};
#pragma unroll
  for (int kk = 0; kk < M_; kk += 4) {
    v2f av, bv0, bv1;
    av.x = xb[(st + kk + 2 * hi) * C_ + ch];      // A[row][m] = x[b, st+m, ch]
    av.y = xb[(st + kk + 2 * hi + 1) * C_ + ch];
    bv0.x = W[d0 * M_ + kk + 2 * hi];             // B[m][d] = W[d][m]
    bv0.y = W[d0 * M_ + kk + 2 * hi + 1];
    bv1.x = W[d1 * M_ + kk + 2 * hi];
    bv1.y = W[d1 * M_ + kk + 2 * hi + 1];
    acc0 = __builtin_amdgcn_wmma_f32_16x16x4_f32(
        false, av, false, bv0, (short)0, acc0, false, false);
    acc1 = __builtin_amdgcn_wmma_f32_16x16x4_f32(
        false, av, false, bv1, (short)0, acc1, false, false);
  }
  const float bval0 = bias[d0];
  const float bval1 = bias[d1];
#pragma unroll
  for (int r = 0; r < 8; ++r) {
    const int orow = rt * 16 + r + 8 * hi;  // C/D layout: M = r + 8*hi
    const int oc = orow >> 3;
    const int ok = orow & 7;
    const size_t base = (((size_t)b * C_ + oc) * K_ + ok) * D_;
    out[base + d0] = acc0[r] + bval0;
    out[base + d1] = acc1[r] + bval1;
  }
}

// ---------- launch ----------
extern "C" void kernel_launch(void* const* d_in, const int* in_sizes, int n_in,
                              void* d_out, int out_size, void* d_ws,
                              size_t ws_size, hipStream_t stream) {
  (void)in_sizes; (void)n_in; (void)out_size; (void)ws_size;
  const float* x = (const float*)d_in[0];     // (8,2048,8) f32
  const float* W = (const float*)d_in[1];     // (512,64)   f32
  const float* bias = (const float*)d_in[2];  // (512,)     f32
  float* out = (float*)d_out;                 // (8,8,8,512) f32

  char* ws = (char*)d_ws;
  float* ts = (float*)(ws + 0);                              // 64 KB
  float* mu = (float*)(ws + 65536);                          // 64 KB
  float* sg = (float*)(ws + 131072);                         // 64 KB
  unsigned long long* cand = (unsigned long long*)(ws + 196608);  // 128 KB
  int* starts = (int*)(ws + 327680);                         // 256 B

  k_prep<<<dim3(B_), dim3(256), 0, stream>>>(x, ts, mu, sg);
  k_scamp<<<dim3(NT_ * NT_, B_), dim3(256), 0, stream>>>(ts, mu, sg, cand);
  k_select<<<dim3(B_), dim3(256), 0, stream>>>(cand, starts);
  k_outgemm<<<dim3(16, 4, B_), dim3(32), 0, stream>>>(x, W, bias, starts, out);
}